// Dual_Slef_Attn_43576738185510
// MI455X (gfx1250) — compile-verified
//
#include <hip/hip_runtime.h>

// ---------------------------------------------------------------------------
// Dual self-attention block for MI455X (gfx1250), fp32 end-to-end.
// - All GEMMs: V_WMMA_F32_16X16X4_F32 (wave32, 16x16 tile/wave, K-step 4).
// - All LDS staging: GLOBAL_LOAD_ASYNC_TO_LDS_B128 (ASYNCcnt-tracked DMA),
//   waited with s_wait_asynccnt before the workgroup barrier.
// ---------------------------------------------------------------------------

typedef float v2f __attribute__((ext_vector_type(2)));
typedef float v8f __attribute__((ext_vector_type(8)));

__device__ __forceinline__ v8f wmma4(v2f a, v2f b, v8f c) {
  // 8-arg pattern: (neg_a, A, neg_b, B, c_mod, C, reuse_a, reuse_b)
  return __builtin_amdgcn_wmma_f32_16x16x4_f32(false, a, false, b, (short)0, c,
                                               false, false);
}

// Async 16B global->LDS copy (GV mode). LDS offset = low 32 bits of generic
// pointer. Both addrs 16B aligned.
__device__ __forceinline__ void async_b128(void* lds_dst, const void* gsrc) {
  const unsigned loff = (unsigned)(uintptr_t)lds_dst;
  const unsigned long long g = (unsigned long long)(uintptr_t)gsrc;
  asm volatile("global_load_async_to_lds_b128 %0, %1, off"
               :
               : "v"(loff), "v"(g)
               : "memory");
}

__device__ __forceinline__ void wait_async() {
  asm volatile("s_wait_asynccnt 0x0" ::: "memory");
}

#define Bn 64
#define Cn 64
#define Nn 256
#define EPSc 1e-5f
// expm1(1e-5) in double; K = 1 + Km1 = exp(1e-5) correctly rounded.
#define Km1 1.0000050000166667e-05f

// ============================================================================
// Kernel 1: all eight 1x1-conv projections fused into one 160x64 GEMM per
// batch. Row map: [0,8) ga_q  [8,16) ga_k  [16,48) ga_v  [48,80) ga_lb
//                 [80,88) sa_q [88,96) sa_k [96,128) sa_v [128,160) sa_lb
// q rows pre-scaled by 1/sqrt(8); lb rows pre-scaled by BN gamma*rsqrt(var+eps)
// (affine offset applied in the epilogue), so consumers read finished values.
// x tile is DMA'd to LDS asynchronously while weights are assembled.
// ============================================================================
__global__ __launch_bounds__(256) void proj_kernel(
    const float* __restrict__ x,
    const float* __restrict__ gaq, const float* __restrict__ gak,
    const float* __restrict__ gav, const float* __restrict__ gavb,
    const float* __restrict__ galb, const float* __restrict__ galbbn,
    const float* __restrict__ saq, const float* __restrict__ sak,
    const float* __restrict__ sav, const float* __restrict__ savb,
    const float* __restrict__ salb, const float* __restrict__ salbbn,
    float* __restrict__ P) {
  __shared__ __align__(16) float wl[160][65];  // weights (stride 65, no cfl.)
  __shared__ __align__(16) float xl[64][68];   // x tile; stride 68 = 16B-align

  const int b = blockIdx.y;
  const int n0 = blockIdx.x * 64;
  const int tid = threadIdx.x;
  const float qscale = 0.35355339059327373f;  // 1/sqrt(8)

  // async DMA: x[b][c][n0..n0+64) -> xl, 1024 x 16B chunks
  for (int idx = tid; idx < 64 * 16; idx += 256) {
    const int c = idx >> 4, j4 = (idx & 15) * 4;
    async_b128(&xl[c][j4], x + (size_t)b * Cn * Nn + c * Nn + n0 + j4);
  }
  // meanwhile: assemble + pre-scale the fused weight matrix
  for (int idx = tid; idx < 160 * 64; idx += 256) {
    const int row = idx >> 6, c = idx & 63;
    float w;
    if (row < 8) {
      w = gaq[row * 64 + c] * qscale;
    } else if (row < 16) {
      w = gak[(row - 8) * 64 + c];
    } else if (row < 48) {
      w = gav[(row - 16) * 64 + c];
    } else if (row < 80) {
      const int ch = row - 48;
      const float s = galbbn[ch] * rsqrtf(galbbn[96 + ch] + EPSc);
      w = galb[ch * 64 + c] * s;
    } else if (row < 88) {
      w = saq[(row - 80) * 64 + c] * qscale;
    } else if (row < 96) {
      w = sak[(row - 88) * 64 + c];
    } else if (row < 128) {
      w = sav[(row - 96) * 64 + c];
    } else {
      const int ch = row - 128;
      const float s = salbbn[ch] * rsqrtf(salbbn[96 + ch] + EPSc);
      w = salb[ch * 64 + c] * s;
    }
    wl[row][c] = w;
  }
  wait_async();
  __syncthreads();

  const int w = tid >> 5, lane = tid & 31, half = lane >> 4, l16 = lane & 15;
  for (int t = w; t < 40; t += 8) {  // 10 row-tiles x 4 col-tiles
    const int mt = t >> 2, nt = t & 3;
    v8f acc = {};
    for (int kt = 0; kt < 16; ++kt) {
      const int k0 = kt * 4 + half * 2;
      v2f a, bb;
      a.x = wl[mt * 16 + l16][k0];
      a.y = wl[mt * 16 + l16][k0 + 1];
      bb.x = xl[k0][nt * 16 + l16];
      bb.y = xl[k0 + 1][nt * 16 + l16];
      acc = wmma4(a, bb, acc);
    }
    const int n = n0 + nt * 16 + l16;
#pragma unroll
    for (int r = 0; r < 8; ++r) {
      const int row = mt * 16 + r + half * 8;
      float val = acc[r];
      if (row >= 16 && row < 48) {
        val += gavb[row - 16];
      } else if (row >= 48 && row < 80) {
        const int ch = row - 48;
        const float s = galbbn[ch] * rsqrtf(galbbn[96 + ch] + EPSc);
        val += galbbn[32 + ch] - galbbn[64 + ch] * s;  // beta - mean*s
      } else if (row >= 96 && row < 128) {
        val += savb[row - 96];
      } else if (row >= 128) {
        const int ch = row - 128;
        const float s = salbbn[ch] * rsqrtf(salbbn[96 + ch] + EPSc);
        val += salbbn[32 + ch] - salbbn[64 + ch] * s;
      }
      P[(size_t)b * 160 * Nn + row * Nn + n] = val;
    }
  }
}

// ============================================================================
// Kernel 2/3: attention over a 16-row tile. Scores (16x256) via WMMA K=8,
// softmax (GA) or exact closed-form stroke-sum (SA) in LDS, then split-K
// WMMA attn@v with LDS partial reduction, fused BN + linear-branch add.
// v tile is DMA'd to LDS asynchronously during q staging.
// ============================================================================
template <bool SA, int PBASE, int FOFF>
__global__ __launch_bounds__(256) void attn_kernel(
    const float* __restrict__ P, const int* __restrict__ stroke,
    const float* __restrict__ bnp, float* __restrict__ F) {
  __shared__ __align__(16) float ql[16][9];    // q tile (stride 9 coprime 64)
  __shared__ __align__(16) float sl[16][257];  // scores / attention weights
  __shared__ __align__(16) float vl[32][260];  // v [vc][m]; stride 260 = 16B
  __shared__ __align__(16) float ps[8][16][16];  // split-K partials
  __shared__ __align__(16) float dl[16][16];     // SA: per-row denominators

  const int b = blockIdx.y;
  const int n0 = blockIdx.x * 16;
  const int tid = threadIdx.x;
  const float* Pb = P + (size_t)b * 160 * Nn;
  const float* Pq = Pb + PBASE * Nn;          // already scaled by 1/sqrt(8)
  const float* Pk = Pb + (PBASE + 8) * Nn;
  const float* Pv = Pb + (PBASE + 16) * Nn;   // bias already folded in
  const float* Plb = Pb + (PBASE + 48) * Nn;  // BN already folded in

  // async DMA: v projection [32][256] -> vl, 2048 x 16B chunks
  for (int idx = tid; idx < 32 * 64; idx += 256) {
    const int vc = idx >> 6, m4 = (idx & 63) * 4;
    async_b128(&vl[vc][m4], Pv + vc * Nn + m4);
  }
  // meanwhile: stage transposed q tile
  for (int idx = tid; idx < 16 * 8; idx += 256) {
    const int n = idx & 15, o = idx >> 4;
    ql[n][o] = Pq[o * Nn + n0 + n];
  }
  wait_async();
  __syncthreads();

  const int w = tid >> 5, lane = tid & 31, half = lane >> 4, l16 = lane & 15;

  // ---- scores: S[16][256] = q[16x8] @ k[8x256] ----
  for (int mt = w; mt < 16; mt += 8) {
    v8f acc = {};
#pragma unroll
    for (int kt = 0; kt < 2; ++kt) {
      const int k0 = kt * 4 + half * 2;
      v2f a, bb;
      a.x = ql[l16][k0];
      a.y = ql[l16][k0 + 1];
      bb.x = Pk[k0 * Nn + mt * 16 + l16];
      bb.y = Pk[(k0 + 1) * Nn + mt * 16 + l16];
      acc = wmma4(a, bb, acc);
    }
#pragma unroll
    for (int r = 0; r < 8; ++r) sl[r + half * 8][mt * 16 + l16] = acc[r];
  }
  __syncthreads();

  // ---- softmax / stroke-sum: 16 threads per row, shfl reductions width 16 --
  {
    const int row = tid >> 4;  // lanes 0-15 / 16-31 of a wave == one row each
    const int j = tid & 15;
    float vals[16];
    float mx = -1e30f;
#pragma unroll
    for (int t = 0; t < 16; ++t) {
      float v = sl[row][j + t * 16];
      if (SA) v *= 10.0f;  // energy = (base+mask)*10
      vals[t] = v;
      mx = fmaxf(mx, v);
    }
#pragma unroll
    for (int d = 1; d < 16; d <<= 1) mx = fmaxf(mx, __shfl_xor(mx, d, 16));
    float sum = 0.f;
#pragma unroll
    for (int t = 0; t < 16; ++t) {
      const float e = __expf(vals[t] - mx);
      vals[t] = e;
      sum += e;
    }
#pragma unroll
    for (int d = 1; d < 16; d <<= 1) sum += __shfl_xor(sum, d, 16);

    if constexpr (!SA) {
      const float inv = 1.0f / sum;
#pragma unroll
      for (int t = 0; t < 16; ++t) sl[row][j + t * 16] = vals[t] * inv;
    } else {
      // Exact collapse of sum_s softmax(10*(base+mask_s)):
      //   D_s = K*T - (K-1)*T_s ; coef[m] = K*R + (1-K)/D_{stroke[m]}
      int st[16];
#pragma unroll
      for (int t = 0; t < 16; ++t) st[t] = stroke[b * Nn + j + t * 16];
      const float Kc = 1.0f + Km1;
      float R = 0.f;
      for (int s = 0; s < 16; ++s) {
        float ts = 0.f;
#pragma unroll
        for (int t = 0; t < 16; ++t) ts += (st[t] == s) ? vals[t] : 0.f;
#pragma unroll
        for (int d = 1; d < 16; d <<= 1) ts += __shfl_xor(ts, d, 16);
        const float D = Kc * sum - Km1 * ts;
        R += 1.0f / D;
        if (j == s) dl[row][s] = D;  // same half-wave produces & consumes
      }
#pragma unroll
      for (int t = 0; t < 16; ++t) {
        const float coef = Kc * R - Km1 / dl[row][st[t]];
        sl[row][j + t * 16] = vals[t] * coef;
      }
    }
  }
  __syncthreads();

  // ---- out[16][32] = attn[16][256] @ v[256][32], split-K over 4 waves ----
  {
    const int vcT = w & 1, kq = w >> 1;
    v8f acc = {};
    for (int kt = 0; kt < 16; ++kt) {
      const int k0 = kq * 64 + kt * 4 + half * 2;
      v2f a, bb;
      a.x = sl[l16][k0];
      a.y = sl[l16][k0 + 1];
      bb.x = vl[vcT * 16 + l16][k0];
      bb.y = vl[vcT * 16 + l16][k0 + 1];
      acc = wmma4(a, bb, acc);
    }
#pragma unroll
    for (int r = 0; r < 8; ++r) ps[w][r + half * 8][l16] = acc[r];
  }
  __syncthreads();

  // ---- reduce split-K, BN, add (pre-BN'd) linear branch, store F ----
  for (int e = tid; e < 512; e += 256) {
    const int vc = e & 31, n = e >> 5, vcT = vc >> 4, c16 = vc & 15;
    float acc = ps[vcT][n][c16] + ps[vcT + 2][n][c16] + ps[vcT + 4][n][c16] +
                ps[vcT + 6][n][c16];
    const float s = bnp[vc] * rsqrtf(bnp[96 + vc] + EPSc);
    const float val =
        (acc - bnp[64 + vc]) * s + bnp[32 + vc] + Plb[vc * Nn + n0 + n];
    F[(size_t)b * Cn * Nn + (FOFF + vc) * Nn + n0 + n] = val;
  }
}

// ============================================================================
// Kernel 4: MLP. h = relu(BN1(W1 @ F)) kept in LDS, out = relu(BN2(W2@h) + F).
// F tile DMA'd to LDS; weights streamed from global as 8B v2f loads, with
// global_prefetch warming before the K-loops.
// ============================================================================
__global__ __launch_bounds__(128) void mlp_kernel(
    const float* __restrict__ F, const float* __restrict__ w1,
    const float* __restrict__ bn1, const float* __restrict__ w2,
    const float* __restrict__ bn2, float* __restrict__ out) {
  __shared__ __align__(16) float fl[64][20];   // stride 20 = 16B aligned rows
  __shared__ __align__(16) float hl[256][17];

  const int b = blockIdx.y;
  const int n0 = blockIdx.x * 16;
  const int tid = threadIdx.x;
  const float* Fb = F + (size_t)b * Cn * Nn;

  // async DMA: F[64][n0..n0+16) -> fl, 256 x 16B chunks
  for (int idx = tid; idx < 64 * 4; idx += 128) {
    const int c = idx >> 2, j4 = (idx & 3) * 4;
    async_b128(&fl[c][j4], Fb + c * Nn + n0 + j4);
  }
  // warm W1/W2 into cache while the DMA runs (16384 floats each)
  __builtin_prefetch(w1 + tid * 128, 0, 0);
  __builtin_prefetch(w2 + tid * 128, 0, 0);
  wait_async();
  __syncthreads();

  const int w = tid >> 5, lane = tid & 31, half = lane >> 4, l16 = lane & 15;

  // ---- h[256][16] = W1[256x64] @ f[64x16] ----
  for (int ot = w; ot < 16; ot += 4) {
    v8f acc = {};
    const int orow = ot * 16 + l16;
    for (int kt = 0; kt < 16; ++kt) {
      const int k0 = kt * 4 + half * 2;
      const v2f a = *reinterpret_cast<const v2f*>(w1 + orow * 64 + k0);
      v2f bb;
      bb.x = fl[k0][l16];
      bb.y = fl[k0 + 1][l16];
      acc = wmma4(a, bb, acc);
    }
#pragma unroll
    for (int r = 0; r < 8; ++r) {
      const int o = ot * 16 + r + half * 8;
      const float s = bn1[o] * rsqrtf(bn1[768 + o] + EPSc);
      const float val = (acc[r] - bn1[512 + o]) * s + bn1[256 + o];
      hl[o][l16] = fmaxf(val, 0.0f);
    }
  }
  __syncthreads();

  // ---- out[64][16] = W2[64x256] @ h[256x16], + residual ----
  {
    const int o0 = w * 16;
    v8f acc = {};
    for (int kt = 0; kt < 64; ++kt) {
      const int k0 = kt * 4 + half * 2;
      const v2f a = *reinterpret_cast<const v2f*>(w2 + (o0 + l16) * 256 + k0);
      v2f bb;
      bb.x = hl[k0][l16];
      bb.y = hl[k0 + 1][l16];
      acc = wmma4(a, bb, acc);
    }
#pragma unroll
    for (int r = 0; r < 8; ++r) {
      const int o = o0 + r + half * 8;
      const float s = bn2[o] * rsqrtf(bn2[192 + o] + EPSc);
      float val = (acc[r] - bn2[128 + o]) * s + bn2[64 + o] + fl[o][l16];
      out[(size_t)b * Cn * Nn + o * Nn + n0 + l16] = fmaxf(val, 0.0f);
    }
  }
}

// ============================================================================
extern "C" void kernel_launch(void* const* d_in, const int* in_sizes, int n_in,
                              void* d_out, int out_size, void* d_ws,
                              size_t ws_size, hipStream_t stream) {
  (void)in_sizes; (void)n_in; (void)out_size; (void)ws_size;
  const float* x      = (const float*)d_in[0];
  const int*   stroke = (const int*)d_in[1];
  const float* gaq    = (const float*)d_in[3];
  const float* gak    = (const float*)d_in[4];
  const float* gav    = (const float*)d_in[5];
  const float* gavb   = (const float*)d_in[6];
  const float* gabn   = (const float*)d_in[7];
  const float* galb   = (const float*)d_in[8];
  const float* galbbn = (const float*)d_in[9];
  const float* saq    = (const float*)d_in[10];
  const float* sak    = (const float*)d_in[11];
  const float* sav    = (const float*)d_in[12];
  const float* savb   = (const float*)d_in[13];
  const float* sabn   = (const float*)d_in[14];
  const float* salb   = (const float*)d_in[15];
  const float* salbbn = (const float*)d_in[16];
  const float* w1     = (const float*)d_in[17];
  const float* bn1    = (const float*)d_in[18];
  const float* w2     = (const float*)d_in[19];
  const float* bn2    = (const float*)d_in[20];

  float* P = (float*)d_ws;               // [64][160][256] = 10.5 MB
  float* F = P + (size_t)Bn * 160 * Nn;  // [64][ 64][256] =  4.0 MB

  proj_kernel<<<dim3(4, Bn), 256, 0, stream>>>(x, gaq, gak, gav, gavb, galb,
                                               galbbn, saq, sak, sav, savb,
                                               salb, salbbn, P);
  attn_kernel<false, 0, 0><<<dim3(16, Bn), 256, 0, stream>>>(P, stroke, gabn,
                                                             F);
  attn_kernel<true, 80, 32><<<dim3(16, Bn), 256, 0, stream>>>(P, stroke, sabn,
                                                              F);
  mlp_kernel<<<dim3(16, Bn), 128, 0, stream>>>(F, w1, bn1, w2, bn2,
                                               (float*)d_out);
}